// dijetReinforceLayer_21818433864076
// MI455X (gfx1250) — compile-verified
//
#include <hip/hip_runtime.h>

// dijetReinforceLayer: out[n,o,s] = sum_{c,k} din[n,c,s,k] * W[o,c,k] + b[o]
// din[n,c,s,0..1] = x[n,c,2s..2s+1], din[n,c,s,2] = d[n,c,s]
// N=131072, ND=32, NPAIR=6 -> batched GEMM M=N*6, N=32, K=96. Memory bound
// (~12 FLOP/byte, ~403 MB moved -> ~17us at 23.3 TB/s), so fp32 WMMA
// 16x16x4 + efficient data movement (async global->LDS scatter) is the play.

#define ND     32
#define NPAIR  6
#define KLEN   96          // 32*3 contraction length
#define PAD    98          // LDS row stride in floats (conflict-free, keeps 8B align)
#define SPW    64          // samples per workgroup
#define ROWS   (SPW*NPAIR) // 384 rows per workgroup
#define THREADS 256        // 8 waves

typedef float v2f __attribute__((ext_vector_type(2)));
typedef float v8f __attribute__((ext_vector_type(8)));

// ---- CDNA5 async global->LDS copies (ASYNCcnt-tracked). GVS addressing:
// vdst = per-lane LDS byte address, vaddr = per-lane 32-bit byte offset,
// saddr = SGPR 64-bit base. Global side stays coalesced; LDS side scatters.
__device__ __forceinline__ void async_g2l_b64(unsigned lds_byte,
                                              const void* sbase,
                                              unsigned voff_byte) {
    asm volatile("global_load_async_to_lds_b64 %0, %1, %2"
                 :: "v"(lds_byte), "v"(voff_byte), "s"(sbase) : "memory");
}
__device__ __forceinline__ void async_g2l_b32(unsigned lds_byte,
                                              const void* sbase,
                                              unsigned voff_byte) {
    asm volatile("global_load_async_to_lds_b32 %0, %1, %2"
                 :: "v"(lds_byte), "v"(voff_byte), "s"(sbase) : "memory");
}
__device__ __forceinline__ void wait_async0() {
    asm volatile("s_wait_asynccnt 0x0" ::: "memory");
}

__global__ __launch_bounds__(THREADS)
void dijet_wmma_f32_kernel(const float* __restrict__ x,
                           const float* __restrict__ dj,
                           const float* __restrict__ W,
                           const float* __restrict__ bias,
                           float* __restrict__ out)
{
    // Staged interleaved rows. Column index u: x-part u = 2*c + k (k=0,1),
    // d-part u = 64 + c. Row stride PAD=98: even (8B-aligned b64 ops) and
    // bank-conflict-free for the 16-row A-fragment gather (98 = 34 mod 64,
    // cross-half collision would need dlane = 17 mod 32 -> dlane = -15: lanes 0 and 31 share banks).
    __shared__ float din[ROWS * PAD];

    const int tid  = threadIdx.x;
    const int lane = tid & 31;
    const int wave = tid >> 5;       // 0..7
    const int lo   = lane & 15;      // N-column within 16
    const int half = lane >> 4;      // 0/1 -> K offset 0/2, M offset 0/8

    const int base_n = blockIdx.x * SPW;

    // ---------------- stage x: async b64, coalesced reads, LDS scatter ------
    {
        const float* xbase = x + (size_t)base_n * (ND * 2 * NPAIR);
        const int n4 = (SPW * ND * 2 * NPAIR) / 4;   // 6144 float4-groups
        for (int i4 = tid; i4 < n4; i4 += THREADS) {
            int i   = i4 * 4;
            int ls  = i / (ND * 2 * NPAIR);           // local sample
            int rem = i - ls * (ND * 2 * NPAIR);
            int c   = rem / (2 * NPAIR);
            int j   = rem - c * (2 * NPAIR);          // 0,4,8
            int s   = j >> 1;                         // 0,2,4
            int dst = (ls * NPAIR + s) * PAD + 2 * c; // even -> 8B aligned
            unsigned l0 = (unsigned)(uintptr_t)&din[dst];
            unsigned l1 = (unsigned)(uintptr_t)&din[dst + PAD];
            async_g2l_b64(l0, xbase, (unsigned)(i * 4));      // (s,   k=0..1)
            async_g2l_b64(l1, xbase, (unsigned)(i * 4 + 8));  // (s+1, k=0..1)
        }
    }
    // ---------------- stage d: async b32 scatter ----------------------------
    {
        const float* dbase = dj + (size_t)base_n * (ND * NPAIR);
        const int n4 = (SPW * ND * NPAIR) / 4;       // 3072 float4-groups
        for (int i4 = tid; i4 < n4; i4 += THREADS) {
            int i = i4 * 4;
            #pragma unroll
            for (int e = 0; e < 4; ++e) {
                int ii  = i + e;
                int ls  = ii / (ND * NPAIR);
                int rem = ii - ls * (ND * NPAIR);
                int c   = rem / NPAIR;
                int s   = rem - c * NPAIR;
                unsigned ld = (unsigned)(uintptr_t)
                              &din[(ls * NPAIR + s) * PAD + 64 + c];
                async_g2l_b32(ld, dbase, (unsigned)(ii * 4));
            }
        }
    }
    wait_async0();        // our async writes landed in LDS
    __syncthreads();      // everyone's writes visible to everyone

    // ---------------- preload W fragments (B matrices) ----------------------
    // B 4x16 f32 layout symmetric to A: lane -> N=lo, VGPR v -> K=v+2*half.
    // LDS column u maps to W offset: u<64: (u>>1)*3 + (u&1) ; u>=64: (u-64)*3+2.
    v2f bf[24][2];
    #pragma unroll
    for (int t = 0; t < 24; ++t) {
        #pragma unroll
        for (int cc = 0; cc < 2; ++cc) {
            int o = cc * 16 + lo;
            #pragma unroll
            for (int v = 0; v < 2; ++v) {
                int p = 4 * t + v + 2 * half;
                int woff;
                if (t < 16) woff = (p >> 1) * 3 + (p & 1);   // x-part
                else        woff = (p - 64) * 3 + 2;         // d-part
                bf[t][cc][v] = W[o * KLEN + woff];
            }
        }
    }
    const float b0 = bias[lo];
    const float b1 = bias[16 + lo];

    // ---------------- compute: 3 row-tiles per wave, 2 col-tiles each -------
    const int wave_row0 = wave * 48;    // 48 rows == 8 whole samples per wave
    #pragma unroll
    for (int tt = 0; tt < 3; ++tt) {
        const int rbase = wave_row0 + tt * 16;
        v8f acc0 = {};
        v8f acc1 = {};
        // A fragment: lane -> M = lo, VGPR v -> K = v + 2*half
        const float* arow = &din[(rbase + lo) * PAD + 2 * half];
        #pragma unroll
        for (int t = 0; t < 24; ++t) {
            v2f a = *(const v2f*)(arow + 4 * t);  // ds_load_b64 (pairs fuse to 2addr)
            acc0 = __builtin_amdgcn_wmma_f32_16x16x4_f32(
                       false, a, false, bf[t][0], (short)0, acc0, false, false);
            acc1 = __builtin_amdgcn_wmma_f32_16x16x4_f32(
                       false, a, false, bf[t][1], (short)0, acc1, false, false);
        }
        // D layout: VGPR v -> local row rbase + 8*half + v, N col = lo
        #pragma unroll
        for (int v = 0; v < 8; ++v) {
            int lr   = rbase + 8 * half + v;      // local row in WG
            int nloc = lr / NPAIR;
            int s    = lr - nloc * NPAIR;
            size_t o0 = ((size_t)(base_n + nloc) * ND + lo) * NPAIR + s;
            out[o0]              = acc0[v] + b0;  // o = lo
            out[o0 + 16 * NPAIR] = acc1[v] + b1;  // o = lo + 16
        }
    }
}

extern "C" void kernel_launch(void* const* d_in, const int* in_sizes, int n_in,
                              void* d_out, int out_size, void* d_ws, size_t ws_size,
                              hipStream_t stream) {
    (void)n_in; (void)out_size; (void)d_ws; (void)ws_size;
    const float* x  = (const float*)d_in[0];
    const float* dj = (const float*)d_in[1];
    const float* W  = (const float*)d_in[2];
    const float* b  = (const float*)d_in[3];
    float* out = (float*)d_out;

    int n = in_sizes[0] / (ND * 2 * NPAIR);   // 131072
    int blocks = n / SPW;                     // 2048
    dijet_wmma_f32_kernel<<<dim3(blocks), dim3(THREADS), 0, stream>>>(x, dj, W, b, out);
}